// LConv_26199300505665
// MI455X (gfx1250) — compile-verified
//
#include <hip/hip_runtime.h>
#include <hip/hip_bf16.h>

// Problem constants (from reference):
//   DIMS=(8,8,16,16) -> X=16384, BATCH=4, NC=3 (complex 3x3 = 18 floats),
//   N_IN=4, N_OUT=8, T_W_SIZE=37.  Output = (u passthrough, w_out).
#define NB      4
#define NX      16384          // 8*8*16*16
#define NSITE   (NB * NX)      // 65536
#define NIN     4
#define NOUT    8
#define NV      37             // t_w channels
#define KPAD    40             // NV padded to multiple of 4 for WMMA K-steps
#define SPW     16             // sites per workgroup
#define COLS    (SPW * 18)     // 288 columns per workgroup tile
#define LSTRIDE 292            // padded LDS row stride (floats)

typedef float v2f __attribute__((ext_vector_type(2)));
typedef float v8f __attribute__((ext_vector_type(8)));

// ---- complex 3x3 helpers; layout m = (i*3+j)*2 + c (c: 0=Re, 1=Im) ----

// C = A * B
__device__ __forceinline__ void mulAB(const float* A, const float* B, float* C) {
#pragma unroll
  for (int i = 0; i < 3; ++i)
#pragma unroll
    for (int k = 0; k < 3; ++k) {
      float cR = 0.f, cI = 0.f;
#pragma unroll
      for (int j = 0; j < 3; ++j) {
        float aR = A[(i * 3 + j) * 2], aI = A[(i * 3 + j) * 2 + 1];
        float bR = B[(j * 3 + k) * 2], bI = B[(j * 3 + k) * 2 + 1];
        cR += aR * bR - aI * bI;
        cI += aR * bI + aI * bR;
      }
      C[(i * 3 + k) * 2] = cR;
      C[(i * 3 + k) * 2 + 1] = cI;
    }
}

// C = A * B^dagger   (C_ik = sum_j A_ij * conj(B_kj))
__device__ __forceinline__ void mulABh(const float* A, const float* B, float* C) {
#pragma unroll
  for (int i = 0; i < 3; ++i)
#pragma unroll
    for (int k = 0; k < 3; ++k) {
      float cR = 0.f, cI = 0.f;
#pragma unroll
      for (int j = 0; j < 3; ++j) {
        float aR = A[(i * 3 + j) * 2], aI = A[(i * 3 + j) * 2 + 1];
        float bR = B[(k * 3 + j) * 2], bI = -B[(k * 3 + j) * 2 + 1];
        cR += aR * bR - aI * bI;
        cI += aR * bI + aI * bR;
      }
      C[(i * 3 + k) * 2] = cR;
      C[(i * 3 + k) * 2 + 1] = cI;
    }
}

// C = A^dagger * B   (C_ik = sum_j conj(A_ji) * B_jk)
__device__ __forceinline__ void mulAhB(const float* A, const float* B, float* C) {
#pragma unroll
  for (int i = 0; i < 3; ++i)
#pragma unroll
    for (int k = 0; k < 3; ++k) {
      float cR = 0.f, cI = 0.f;
#pragma unroll
      for (int j = 0; j < 3; ++j) {
        float aR = A[(j * 3 + i) * 2], aI = -A[(j * 3 + i) * 2 + 1];
        float bR = B[(j * 3 + k) * 2], bI = B[(j * 3 + k) * 2 + 1];
        cR += aR * bR - aI * bI;
        cI += aR * bI + aI * bR;
      }
      C[(i * 3 + k) * 2] = cR;
      C[(i * 3 + k) * 2 + 1] = cI;
    }
}

__global__ __launch_bounds__(256) void lconv_kernel(
    const float* __restrict__ u, const float* __restrict__ w,
    const float* __restrict__ weight, float* __restrict__ wout) {
  // TW tile: rows = t_w channel v (0..39, 37..39 zero-pad), cols = site_local*18 + m
  __shared__ float TWs[KPAD][LSTRIDE];

  const int tid = threadIdx.x;
  const int s0 = blockIdx.x * SPW;  // first global site of this workgroup

  // ---- Phase 1a: rows 0..3 = raw w channels (contiguous 72 floats / site) ----
  for (int i = tid; i < SPW * NIN * 18; i += 256) {
    int site = i / 72;
    int r = (i / 18) & 3;
    int m = i % 18;
    TWs[r][site * 18 + m] = w[(size_t)(s0 + site) * 72 + r * 18 + m];
  }

  // ---- Phase 1b: row 36 = identity term, rows 37..39 = zero padding ----
  for (int i = tid; i < 4 * COLS; i += 256) {
    int r = 36 + i / COLS;
    int col = i % COLS;
    int m = col % 18;
    TWs[r][col] = (r == 36 && (m == 0 || m == 8 || m == 16)) ? 1.f : 0.f;
  }

  // ---- Phase 1c: rows 4..35 = transported terms (VALU complex sandwiches) ----
  // 16 sites * 8 (orientation,axis) combos * 4 channels = 512 tasks
  for (int t = tid; t < SPW * 32; t += 256) {
    int site = t >> 5;
    int combo = (t >> 2) & 7;  // 0..3: orientation +1, axis=combo; 4..7: orientation -1
    int n = t & 3;
    int s = s0 + site;
    int b = s >> 14;        // / NX
    int x = s & (NX - 1);
    int axis = combo & 3;
    bool pos = combo < 4;

    // lattice coords (C-order: d0*2048 + d1*256 + d2*16 + d3; dims 8,8,16,16)
    int shamt = (axis == 0) ? 11 : (axis == 1) ? 8 : (axis == 2) ? 4 : 0;
    int dmask = (axis < 2) ? 7 : 15;
    int c = (x >> shamt) & dmask;
    int cn = pos ? ((c + 1) & dmask) : ((c - 1) & dmask);
    int xn = x + ((cn - c) << shamt);  // neighbor site along +/- axis

    // orientation +1: V = U[x]   * W[x+e] * U[x]^dag
    // orientation -1: V = U[x-e]^dag * W[x-e] * U[x-e]
    int xa = pos ? x : xn;  // site where U is read
    const float* Ap = u + ((size_t)(b * NX + xa) * 4 + axis) * 18;
    const float* Wp = w + ((size_t)(b * NX + xn) * 4 + n) * 18;

    float A[18], W[18], T[18], V[18];
#pragma unroll
    for (int m = 0; m < 18; ++m) A[m] = Ap[m];
#pragma unroll
    for (int m = 0; m < 18; ++m) W[m] = Wp[m];

    if (pos) {
      mulAB(A, W, T);
      mulABh(T, A, V);
    } else {
      mulAhB(A, W, T);
      mulAB(T, A, V);
    }

    int row = 4 + combo * 4 + n;
#pragma unroll
    for (int m = 0; m < 18; ++m) TWs[row][site * 18 + m] = V[m];
  }

  __syncthreads();

  // ---- Phase 2: Out[16pad x 288] = weightpad[16 x 40] @ TW[40 x 288] via
  //      V_WMMA_F32_16X16X4_F32, 10 K-steps per 16-column N-tile ----
  const int lane = tid & 31;
  const int wave = tid >> 5;
  const int rrow = lane & 15;        // A-matrix row M (lanes 0-15 and 16-31)
  const int kh = (lane >> 4) * 2;    // K sub-pair: lanes 0-15 -> K+0/1, 16-31 -> K+2/3

  // Preload A fragments (weight, zero-padded to 16 rows x 40 K) for all 10 K-steps
  float ax[10], ay[10];
#pragma unroll
  for (int st = 0; st < 10; ++st) {
    int k0 = st * 4 + kh;
    ax[st] = (rrow < NOUT && k0 < NV) ? weight[rrow * NV + k0] : 0.f;
    ay[st] = (rrow < NOUT && (k0 + 1) < NV) ? weight[rrow * NV + k0 + 1] : 0.f;
  }

  for (int tile = wave; tile < COLS / 16; tile += 8) {  // wave-uniform: EXEC stays full
    int nbase = tile * 16;
    v8f acc = {0.f, 0.f, 0.f, 0.f, 0.f, 0.f, 0.f, 0.f};
#pragma unroll
    for (int st = 0; st < 10; ++st) {
      int k0 = st * 4 + kh;
      v2f a, bf;
      a.x = ax[st];
      a.y = ay[st];
      bf.x = TWs[k0][nbase + rrow];       // B: VGPR0 holds K=k0 (lanes 0-15) / k0+2 (16-31)
      bf.y = TWs[k0 + 1][nbase + rrow];   //    VGPR1 holds K=k0+1 / k0+3
      acc = __builtin_amdgcn_wmma_f32_16x16x4_f32(
          /*neg_a=*/false, a, /*neg_b=*/false, bf,
          /*c_mod=*/(short)0, acc, /*reuse_a=*/false, /*reuse_b=*/false);
    }
    // D layout: VGPR r -> M=r (lanes 0-15), M=r+8 (lanes 16-31). Rows 8..15 are
    // zero padding, so only lanes 0..15 carry real outputs (u_out = r).
    if (lane < 16) {
      int col = nbase + lane;
      int site = col / 18;
      int m = col % 18;
      size_t base = (size_t)(s0 + site) * (NOUT * 18) + m;
#pragma unroll
      for (int r = 0; r < NOUT; ++r) wout[base + r * 18] = acc[r];
    }
  }
}

extern "C" void kernel_launch(void* const* d_in, const int* in_sizes, int n_in,
                              void* d_out, int out_size, void* d_ws, size_t ws_size,
                              hipStream_t stream) {
  const float* u = (const float*)d_in[0];       // (4,16384,4,3,3,2)
  const float* w = (const float*)d_in[1];       // (4,16384,4,3,3,2)
  const float* weight = (const float*)d_in[2];  // (8,37)
  float* out = (float*)d_out;

  const size_t u_elems = (size_t)NB * NX * 4 * 18;  // 4,718,592
  // Output 0: u passthrough (DMA copy on stream; graph-capture safe).
  hipMemcpyAsync(out, u, u_elems * sizeof(float), hipMemcpyDeviceToDevice, stream);

  // Output 1: w_out (4,16384,8,3,3,2) right after u.
  float* wout = out + u_elems;
  lconv_kernel<<<NSITE / SPW, 256, 0, stream>>>(u, w, weight, wout);
}